// PoolingConv_43602507989837
// MI455X (gfx1250) — compile-verified
//
#include <hip/hip_runtime.h>
#include <cstdint>

// Problem constants (match reference).
static constexpr int   NN      = 100000;   // nodes
static constexpr int   NE      = 1600000;  // edges per etype (multiple of 4!)
static constexpr int   DF      = 128;      // feature dim (= 32 lanes * float4)
static constexpr int   TILE    = 1024;     // edges staged per block
static constexpr int   BLOCK   = 256;      // 8 waves (wave32)
static constexpr int   NWAVES  = BLOCK / 32;

// Native CDNA5 f32 atomic add, no-return form (STOREcnt-tracked).
// No memory clobber: volatile preserves execution + mutual order, but lets
// the compiler hoist independent gather loads above it (software pipelining).
__device__ __forceinline__ void atom_add_f32(float* p, float v) {
    asm volatile("global_atomic_add_f32 %0, %1, off" :: "v"(p), "v"(v));
}

// ---------------------------------------------------------------------------
// Edge scatter: for each edge e, sum[dst[e]][:] += feat[src[e]][:],
// cnt[dst[e]] += 1.  One wave per edge; lane l owns floats [4l, 4l+4).
// Edge indices for the whole block are staged into LDS with CDNA5 async
// global->LDS copies (ASYNCcnt).  NE and TILE are multiples of 4, so each
// staging thread's 4 slots are all-valid or all-invalid -> no scalar tail.
// ---------------------------------------------------------------------------
__global__ void __launch_bounds__(BLOCK)
gmp_scatter(const float* __restrict__ feat,
            const int*   __restrict__ src,
            const int*   __restrict__ dst,
            float*       __restrict__ sum,
            float*       __restrict__ cnt,
            int nEdges) {
    __shared__ int s_src[TILE];
    __shared__ int s_dst[TILE];

    const int t        = threadIdx.x;
    const int tileBase = blockIdx.x * TILE;

    // ---- stage TILE src + TILE dst indices into LDS (16B/lane async) ----
    {
        const int base = tileBase + t * 4;
        if (base + 3 < nEdges) {              // all-or-nothing per 16B granule
            uint32_t lsrc = (uint32_t)(uintptr_t)(&s_src[t * 4]);
            uint32_t ldst = (uint32_t)(uintptr_t)(&s_dst[t * 4]);
            const int* gs = src + base;
            const int* gd = dst + base;
            asm volatile("global_load_async_to_lds_b128 %0, %1, off"
                         :: "v"(lsrc), "v"(gs) : "memory");
            asm volatile("global_load_async_to_lds_b128 %0, %1, off"
                         :: "v"(ldst), "v"(gd) : "memory");
        }
        asm volatile("s_wait_asynccnt 0" ::: "memory");
        __syncthreads();
    }

    const int lane  = t & 31;
    const int wave  = t >> 5;
    const int rem   = nEdges - tileBase;
    const int limit = rem < TILE ? rem : TILE;   // uniform trip bound, no break

#pragma unroll 2
    for (int i = wave; i < limit; i += NWAVES) {
        // wave-uniform indices -> scalar regs so gathers use saddr form
        const int s = __builtin_amdgcn_readfirstlane(s_src[i]);
        const int d = __builtin_amdgcn_readfirstlane(s_dst[i]);

        // prefetch the feature row of this wave's next edge (hides L2 latency)
        const int i2 = i + NWAVES;
        if (i2 < limit) {
            const int sn = s_src[i2];
            __builtin_prefetch(feat + (size_t)sn * DF + lane * 4, 0, 0);
        }

        const float4 v = *(const float4*)(feat + (size_t)s * DF + lane * 4);
        float* o = sum + (size_t)d * DF + lane * 4;
        atom_add_f32(o + 0, v.x);
        atom_add_f32(o + 1, v.y);
        atom_add_f32(o + 2, v.z);
        atom_add_f32(o + 3, v.w);
        if (lane == 0) atom_add_f32(cnt + d, 1.0f);
    }
}

// ---------------------------------------------------------------------------
// Zero a float4 region.
// ---------------------------------------------------------------------------
__global__ void __launch_bounds__(BLOCK)
gmp_zero4(float4* __restrict__ p, long n4) {
    long i = (long)blockIdx.x * blockDim.x + threadIdx.x;
    if (i < n4) p[i] = make_float4(0.f, 0.f, 0.f, 0.f);
}

// ---------------------------------------------------------------------------
// out = feat + 0.5*(sum0/c0)*[c0>0] + 0.5*(sum1/c1)*[c1>0]
// sum0 lives in `out` (in-place read-modify-write, thread-private elements).
// ---------------------------------------------------------------------------
__global__ void __launch_bounds__(BLOCK)
gmp_finalize(const float* __restrict__ feat,
             float*       __restrict__ out,    // holds sum0
             const float* __restrict__ sum1,
             const float* __restrict__ cnt0,
             const float* __restrict__ cnt1) {
    const long total4 = (long)NN * (DF / 4);
    long i4 = (long)blockIdx.x * blockDim.x + threadIdx.x;
    if (i4 >= total4) return;
    const int n = (int)(i4 >> 5);                  // DF/4 == 32 float4 per node

    const float c0 = cnt0[n];
    const float c1 = cnt1[n];
    const float r0 = (c0 > 0.f) ? 0.5f / c0 : 0.f;
    const float r1 = (c1 > 0.f) ? 0.5f / c1 : 0.f;

    const float4 f  = ((const float4*)feat)[i4];
    const float4 s0 = ((const float4*)out )[i4];
    const float4 s1 = ((const float4*)sum1)[i4];

    float4 r;
    r.x = fmaf(s0.x, r0, fmaf(s1.x, r1, f.x));
    r.y = fmaf(s0.y, r0, fmaf(s1.y, r1, f.y));
    r.z = fmaf(s0.z, r0, fmaf(s1.z, r1, f.z));
    r.w = fmaf(s0.w, r0, fmaf(s1.w, r1, f.w));
    ((float4*)out)[i4] = r;
}

// ---------------------------------------------------------------------------
// Launch wrapper.
//   d_in: feat, src0, dst0, src1, dst1
//   d_out: N*D floats (used first as sum0 accumulator, then final output)
//   d_ws : sum1 (N*D f32) | cnt0 (N f32) | cnt1 (N f32)  ~= 52 MB
// ---------------------------------------------------------------------------
extern "C" void kernel_launch(void* const* d_in, const int* in_sizes, int n_in,
                              void* d_out, int out_size, void* d_ws, size_t ws_size,
                              hipStream_t stream) {
    (void)in_sizes; (void)n_in; (void)out_size; (void)ws_size;

    const float* feat = (const float*)d_in[0];
    const int*   src0 = (const int*)d_in[1];
    const int*   dst0 = (const int*)d_in[2];
    const int*   src1 = (const int*)d_in[3];
    const int*   dst1 = (const int*)d_in[4];

    float* sum0 = (float*)d_out;
    float* ws   = (float*)d_ws;
    float* sum1 = ws;
    float* cnt0 = ws + (size_t)NN * DF;
    float* cnt1 = cnt0 + NN;

    // zero sum0 (d_out) and the contiguous ws region (sum1|cnt0|cnt1)
    {
        const long n4a = (long)NN * DF / 4;
        const long n4b = ((long)NN * DF + 2L * NN) / 4;
        gmp_zero4<<<(int)((n4a + BLOCK - 1) / BLOCK), BLOCK, 0, stream>>>((float4*)sum0, n4a);
        gmp_zero4<<<(int)((n4b + BLOCK - 1) / BLOCK), BLOCK, 0, stream>>>((float4*)ws,   n4b);
    }

    const int tiles = (NE + TILE - 1) / TILE;
    gmp_scatter<<<tiles, BLOCK, 0, stream>>>(feat, src0, dst0, sum0, cnt0, NE);
    gmp_scatter<<<tiles, BLOCK, 0, stream>>>(feat, src1, dst1, sum1, cnt1, NE);

    const long total4 = (long)NN * (DF / 4);
    gmp_finalize<<<(int)((total4 + BLOCK - 1) / BLOCK), BLOCK, 0, stream>>>(
        feat, sum0, sum1, cnt0, cnt1);
}